// AlignmentModel_36756330119363
// MI455X (gfx1250) — compile-verified
//
#include <hip/hip_runtime.h>
#include <math.h>

// ---------------------------------------------------------------------------
// AlignmentModel fused pipeline for gfx1250 (MI455X).
//   K1: T1  = text  @ W1        (512x768x768)  f32 WMMA, NN
//   K2: C   = tanh(T1 @ visual^T) (512x512x768) f32 WMMA, NT, tanh epilogue
//   K3: W2t = text  @ W2        (512x768x768)  f32 WMMA, NN
//   K4: W3v = visual@ W3^T      (512x768x768)  f32 WMMA, NT
//   K5: score[n,m] = sum_d text[n,d]*tanh(W2t[n,d] + W3v[m,d]*C[n,m])
// All intermediates live in L2 (~6MB << 192MB); K5 is VALU/tanh bound.
// ---------------------------------------------------------------------------

typedef __attribute__((ext_vector_type(2))) float v2f;
typedef __attribute__((ext_vector_type(8))) float v8f;

#define D_DIM 768
#define N_ROWS 512
#define M_ROWS 512

__device__ __forceinline__ float fast_tanh(float x) {
#if __has_builtin(__builtin_amdgcn_tanhf)
  return __builtin_amdgcn_tanhf(x);   // v_tanh_f32 on gfx1250
#else
  return tanhf(x);
#endif
}

// ---------------------------------------------------------------------------
// f32 GEMM via V_WMMA_F32_16X16X4_F32. One wave computes a 16x64 strip of C.
// BTRANS=0: B element (k,n) = Bp[k*ldb + n]   (row-major KxN)
// BTRANS=1: B element (k,n) = Bp[n*ldb + k]   (C = A @ W^T, W row-major NxK)
// DO_TANH: apply tanh to the accumulator before store.
//
// Fragment layouts (ISA 7.12.2, 32-bit):
//   A 16x4 : lane l(0..15),half h: {A[row0+l][k+2h], A[row0+l][k+2h+1]}
//   B 4x16 : VGPR0 = rows k+2h, VGPR1 = rows k+2h+1, cols striped on lanes
//   C 16x16: VGPR v -> row (v + 8h), col = col0 + l
// ---------------------------------------------------------------------------
template <int BTRANS, int DO_TANH>
__global__ void gemm_wmma_f32(const float* __restrict__ A,
                              const float* __restrict__ B,
                              float* __restrict__ C,
                              int Mrows, int Ncols, int K,
                              int lda, int ldb, int ldc) {
  const int lane = threadIdx.x & 31;
  const int wave = threadIdx.x >> 5;
  const int tilesX = Ncols >> 6;                       // 64-wide strips
  const int tid = blockIdx.x * (blockDim.x >> 5) + wave;
  const int trow = tid / tilesX;
  const int tcol = tid - trow * tilesX;
  if (trow >= (Mrows >> 4)) return;                    // wave-uniform guard
  const int row0 = trow << 4;
  const int col0 = tcol << 6;

  const int h = lane >> 4;                             // lane half (0/1)
  const int l = lane & 15;

  v8f acc0 = {}, acc1 = {}, acc2 = {}, acc3 = {};
  const float* Arow = A + (size_t)(row0 + l) * lda + 2 * h;

#pragma unroll 2
  for (int k = 0; k < K; k += 4) {
    v2f a = *(const v2f*)(Arow + k);
    v2f b0, b1, b2, b3;
    if (BTRANS) {
      const float* Bb = B + (size_t)k + 2 * h;
      b0 = *(const v2f*)(Bb + (size_t)(col0 + l)      * ldb);
      b1 = *(const v2f*)(Bb + (size_t)(col0 + 16 + l) * ldb);
      b2 = *(const v2f*)(Bb + (size_t)(col0 + 32 + l) * ldb);
      b3 = *(const v2f*)(Bb + (size_t)(col0 + 48 + l) * ldb);
    } else {
      const float* Br0 = B + (size_t)(k + 2 * h) * ldb + col0 + l;
      const float* Br1 = Br0 + ldb;
      b0.x = Br0[0];  b0.y = Br1[0];
      b1.x = Br0[16]; b1.y = Br1[16];
      b2.x = Br0[32]; b2.y = Br1[32];
      b3.x = Br0[48]; b3.y = Br1[48];
    }
    acc0 = __builtin_amdgcn_wmma_f32_16x16x4_f32(false, a, false, b0,
                                                 (short)0, acc0, false, false);
    acc1 = __builtin_amdgcn_wmma_f32_16x16x4_f32(false, a, false, b1,
                                                 (short)0, acc1, false, false);
    acc2 = __builtin_amdgcn_wmma_f32_16x16x4_f32(false, a, false, b2,
                                                 (short)0, acc2, false, false);
    acc3 = __builtin_amdgcn_wmma_f32_16x16x4_f32(false, a, false, b3,
                                                 (short)0, acc3, false, false);
  }

  float* Cr = C + (size_t)row0 * ldc + col0;
#pragma unroll
  for (int v = 0; v < 8; ++v) {
    const int r = v + 8 * h;
    float x0 = acc0[v], x1 = acc1[v], x2 = acc2[v], x3 = acc3[v];
    if (DO_TANH) {
      x0 = fast_tanh(x0); x1 = fast_tanh(x1);
      x2 = fast_tanh(x2); x3 = fast_tanh(x3);
    }
    float* Crr = Cr + (size_t)r * ldc + l;
    Crr[0] = x0; Crr[16] = x1; Crr[32] = x2; Crr[48] = x3;
  }
}

// ---------------------------------------------------------------------------
// K5: one wave computes 4 adjacent scores for one row n.
// score[n,m] = sum_d text[n,d] * tanh(W2t[n,d] + W3v[m,d] * C[n,m])
// ---------------------------------------------------------------------------
__global__ void fused_score(const float* __restrict__ text,
                            const float* __restrict__ w2t,
                            const float* __restrict__ w3v,
                            const float* __restrict__ Cm,
                            float* __restrict__ out) {
  const int lane = threadIdx.x & 31;
  const int wid = (blockIdx.x * blockDim.x + threadIdx.x) >> 5;
  const int n  = wid >> 7;            // / (512/4)
  const int m0 = (wid & 127) << 2;    // 4 columns per wave

  const float* tR = text + (size_t)n * D_DIM;
  const float* aR = w2t  + (size_t)n * D_DIM;
  const float* bR = w3v  + (size_t)m0 * D_DIM;

  const float4 c4 = *(const float4*)(Cm + (size_t)n * M_ROWS + m0);
  const float c[4] = {c4.x, c4.y, c4.z, c4.w};
  float s[4] = {0.f, 0.f, 0.f, 0.f};

#pragma unroll
  for (int i = 0; i < D_DIM / 128; ++i) {
    const int d = i * 128 + lane * 4;
    const float4 t = *(const float4*)(tR + d);
    const float4 a = *(const float4*)(aR + d);
#pragma unroll
    for (int j = 0; j < 4; ++j) {
      const float4 b = *(const float4*)(bR + (size_t)j * D_DIM + d);
      s[j] += t.x * fast_tanh(fmaf(b.x, c[j], a.x));
      s[j] += t.y * fast_tanh(fmaf(b.y, c[j], a.y));
      s[j] += t.z * fast_tanh(fmaf(b.z, c[j], a.z));
      s[j] += t.w * fast_tanh(fmaf(b.w, c[j], a.w));
    }
  }

#pragma unroll
  for (int j = 0; j < 4; ++j)
#pragma unroll
    for (int off = 16; off > 0; off >>= 1)
      s[j] += __shfl_xor(s[j], off, 32);

  if (lane == 0) {
    float4 o = {s[0], s[1], s[2], s[3]};
    *(float4*)(out + (size_t)n * M_ROWS + m0) = o;
  }
}

extern "C" void kernel_launch(void* const* d_in, const int* in_sizes, int n_in,
                              void* d_out, int out_size, void* d_ws, size_t ws_size,
                              hipStream_t stream) {
  const float* text   = (const float*)d_in[0];   // (512,768)
  const float* visual = (const float*)d_in[1];   // (512,768)
  const float* W1     = (const float*)d_in[2];   // (768,768)
  const float* W2     = (const float*)d_in[3];   // (768,768)
  const float* W3     = (const float*)d_in[4];   // (768,768)
  float* out = (float*)d_out;                    // (512,512)

  float* ws  = (float*)d_ws;
  float* T1  = ws;                                // 512*768
  float* W2t = T1  + N_ROWS * D_DIM;              // 512*768
  float* W3v = W2t + N_ROWS * D_DIM;              // 512*768
  float* Cm  = W3v + M_ROWS * D_DIM;              // 512*512
  (void)in_sizes; (void)n_in; (void)out_size; (void)ws_size;

  const dim3 blk(128);   // 4 waves/block, each wave -> 16x64 output strip

  // K1: T1 = text @ W1            tiles: 32 x 12 = 384 waves -> 96 blocks
  gemm_wmma_f32<0, 0><<<96, blk, 0, stream>>>(text, W1, T1,
                                              N_ROWS, D_DIM, D_DIM,
                                              D_DIM, D_DIM, D_DIM);
  // K3: W2t = text @ W2
  gemm_wmma_f32<0, 0><<<96, blk, 0, stream>>>(text, W2, W2t,
                                              N_ROWS, D_DIM, D_DIM,
                                              D_DIM, D_DIM, D_DIM);
  // K4: W3v = visual @ W3^T       (NT: B(k,d) = W3[d*768 + k])
  gemm_wmma_f32<1, 0><<<96, blk, 0, stream>>>(visual, W3, W3v,
                                              M_ROWS, D_DIM, D_DIM,
                                              D_DIM, D_DIM, D_DIM);
  // K2: Cm = tanh(T1 @ visual^T)  tiles: 32 x 8 = 256 waves -> 64 blocks
  gemm_wmma_f32<1, 1><<<64, blk, 0, stream>>>(T1, visual, Cm,
                                              N_ROWS, M_ROWS, D_DIM,
                                              D_DIM, D_DIM, M_ROWS);
  // K5: fused score. 512 * 128 waves, 8 waves/block -> 8192 blocks.
  fused_score<<<8192, 256, 0, stream>>>(text, W2t, W3v, Cm, out);
}